// MyGRU_91225105367262
// MI455X (gfx1250) — compile-verified
//
#include <hip/hip_runtime.h>
#include <math.h>

#define B_  64
#define E_  512
#define H_  1024
#define T_  512
#define CK_ (H_ + E_)      // 1536

#define NWG    40
#define BLK    128
#define NWAVES (NWG * (BLK / 32))   // 160 waves

typedef __attribute__((ext_vector_type(2))) float v2f;
typedef __attribute__((ext_vector_type(8))) float v8f;

__device__ __forceinline__ float sigmoidf_(float x) {
    return 1.0f / (1.0f + __expf(-x));
}

// One wave computes a 64(M) x 16(N) strip: 4 accumulators, K advances by 4 per WMMA.
// a0..a3: per-lane A pointers (row = 16*j + (lane&15), already offset by 2*(lane>>4)).
// bp:     per-lane B pointer  (row n = n0 + (lane&15) of W, offset by 2*(lane>>4)).
__device__ __forceinline__ void mma_strip(v8f* c,
    const float* a0, const float* a1, const float* a2, const float* a3,
    const float* bp, int quads)
{
#pragma unroll 2
    for (int k = 0; k < quads; ++k) {
        v2f b  = *(const v2f*)bp;
        v2f x0 = *(const v2f*)a0;
        v2f x1 = *(const v2f*)a1;
        v2f x2 = *(const v2f*)a2;
        v2f x3 = *(const v2f*)a3;
        c[0] = __builtin_amdgcn_wmma_f32_16x16x4_f32(false, x0, false, b, (short)0, c[0], false, false);
        c[1] = __builtin_amdgcn_wmma_f32_16x16x4_f32(false, x1, false, b, (short)0, c[1], false, false);
        c[2] = __builtin_amdgcn_wmma_f32_16x16x4_f32(false, x2, false, b, (short)0, c[2], false, false);
        c[3] = __builtin_amdgcn_wmma_f32_16x16x4_f32(false, x3, false, b, (short)0, c[3], false, false);
        bp += 4; a0 += 4; a1 += 4; a2 += 4; a3 += 4;
    }
}

// Device-wide sense-free barrier: monotonically increasing counter, target = epoch*NWG.
__device__ __forceinline__ void grid_barrier(unsigned* bar, unsigned target)
{
    __threadfence();            // publish this block's stores at device scope
    __syncthreads();
    if (threadIdx.x == 0) {
        __hip_atomic_fetch_add(bar, 1u, __ATOMIC_RELEASE, __HIP_MEMORY_SCOPE_AGENT);
        unsigned v;
        do {
            __builtin_amdgcn_s_sleep(2);
            v = __hip_atomic_load(bar, __ATOMIC_ACQUIRE, __HIP_MEMORY_SCOPE_AGENT);
        } while (v < target);
    }
    __syncthreads();
    __threadfence();            // acquire: invalidate stale lines before re-reading
}

__global__ void gru_init_ws(float* h0, unsigned* bar)
{
    int i = blockIdx.x * blockDim.x + threadIdx.x;
    if (i == 0) bar[0] = 0u;
    if (i < B_ * H_) h0[i] = 0.0f;
}

__global__ __launch_bounds__(BLK, 2)
void gru_persistent(const float* __restrict__ word,
                    const float* __restrict__ Wz, const float* __restrict__ bz,
                    const float* __restrict__ Wr, const float* __restrict__ br,
                    const float* __restrict__ Wh, const float* __restrict__ bh,
                    const float* __restrict__ Wy, const float* __restrict__ by,
                    float* __restrict__ out,
                    float* __restrict__ ws, unsigned* __restrict__ bar)
{
    const int lane = threadIdx.x & 31;
    const int wave = blockIdx.x * (BLK / 32) + (threadIdx.x >> 5);
    const int hi   = lane >> 4;      // selects K pair / M+8 half per ISA layout
    const int l15  = lane & 15;
    const int koff = 2 * hi;

    const int BH = B_ * H_;
    float* zb  = ws + 2 * BH;        // z gate    [B, H]
    float* rhb = ws + 3 * BH;        // r*h       [B, H]

    unsigned epoch = 0;

    for (int t = 0; t < T_; ++t) {
        const float* hOld = ws + ((t)     & 1) * BH;   // h entering step t
        float*       hNew = ws + ((t + 1) & 1) * BH;

        // ---------------- Phase 1: z, r (and y of step t-1) ----------------
        for (int task = wave; task < 160; task += NWAVES) {
            if (task < 128) {
                const bool  isZ  = (task < 64);
                const float* W   = isZ ? Wz : Wr;
                const float* bia = isZ ? bz : br;
                const int    n0  = (task & 63) * 16;
                const int    n   = n0 + l15;

                v8f c[4] = {};
                const float* bp = W + (size_t)n * CK_ + koff;
                const float* a0 = hOld + (size_t)( 0 + l15) * H_ + koff;
                const float* a1 = hOld + (size_t)(16 + l15) * H_ + koff;
                const float* a2 = hOld + (size_t)(32 + l15) * H_ + koff;
                const float* a3 = hOld + (size_t)(48 + l15) * H_ + koff;
                mma_strip(c, a0, a1, a2, a3, bp, H_ / 4);          // h part of comb
                if (t > 0) {                                       // x = word[:, t-1]
                    const size_t xoff = (size_t)(t - 1) * E_ + koff;
                    const float* w0 = word + (size_t)( 0 + l15) * (T_ * E_) + xoff;
                    const float* w1 = word + (size_t)(16 + l15) * (T_ * E_) + xoff;
                    const float* w2 = word + (size_t)(32 + l15) * (T_ * E_) + xoff;
                    const float* w3 = word + (size_t)(48 + l15) * (T_ * E_) + xoff;
                    mma_strip(c, w0, w1, w2, w3, bp + H_, E_ / 4); // x part of comb
                }

                const float bn = bia[n];
#pragma unroll
                for (int j = 0; j < 4; ++j) {
#pragma unroll
                    for (int i = 0; i < 8; ++i) {
                        const int m   = 16 * j + i + 8 * hi;
                        const int idx = m * H_ + n;
                        const float g = sigmoidf_(c[j][i] + bn);
                        if (isZ) zb[idx]  = g;
                        else     rhb[idx] = g * hOld[idx];
                    }
                }
            } else if (t > 0) {
                // y_{t-1} = hOld @ Wy^T + by   (hOld is h after step t-1)
                const int n0 = (task - 128) * 16;
                const int n  = n0 + l15;
                v8f c[4] = {};
                const float* bp = Wy + (size_t)n * H_ + koff;
                const float* a0 = hOld + (size_t)( 0 + l15) * H_ + koff;
                const float* a1 = hOld + (size_t)(16 + l15) * H_ + koff;
                const float* a2 = hOld + (size_t)(32 + l15) * H_ + koff;
                const float* a3 = hOld + (size_t)(48 + l15) * H_ + koff;
                mma_strip(c, a0, a1, a2, a3, bp, H_ / 4);
                const float bn = by[n];
#pragma unroll
                for (int j = 0; j < 4; ++j) {
#pragma unroll
                    for (int i = 0; i < 8; ++i) {
                        const int m = 16 * j + i + 8 * hi;
                        out[(size_t)m * (T_ * E_) + (size_t)(t - 1) * E_ + n] = c[j][i] + bn;
                    }
                }
            }
        }
        grid_barrier(bar, (++epoch) * NWG);

        // ---------------- Phase 2: h_tilde, h_new ----------------
        for (int task = wave; task < 64; task += NWAVES) {
            const int n0 = task * 16;
            const int n  = n0 + l15;
            v8f c[4] = {};
            const float* bp = Wh + (size_t)n * CK_ + koff;
            const float* a0 = rhb + (size_t)( 0 + l15) * H_ + koff;
            const float* a1 = rhb + (size_t)(16 + l15) * H_ + koff;
            const float* a2 = rhb + (size_t)(32 + l15) * H_ + koff;
            const float* a3 = rhb + (size_t)(48 + l15) * H_ + koff;
            mma_strip(c, a0, a1, a2, a3, bp, H_ / 4);
            if (t > 0) {
                const size_t xoff = (size_t)(t - 1) * E_ + koff;
                const float* w0 = word + (size_t)( 0 + l15) * (T_ * E_) + xoff;
                const float* w1 = word + (size_t)(16 + l15) * (T_ * E_) + xoff;
                const float* w2 = word + (size_t)(32 + l15) * (T_ * E_) + xoff;
                const float* w3 = word + (size_t)(48 + l15) * (T_ * E_) + xoff;
                mma_strip(c, w0, w1, w2, w3, bp + H_, E_ / 4);
            }
            const float bn = bh[n];
#pragma unroll
            for (int j = 0; j < 4; ++j) {
#pragma unroll
                for (int i = 0; i < 8; ++i) {
                    const int m   = 16 * j + i + 8 * hi;
                    const int idx = m * H_ + n;
                    const float ht = tanhf(c[j][i] + bn);
                    const float zv = zb[idx];
                    const float hp = hOld[idx];
                    hNew[idx] = fmaf(zv, ht - hp, hp);   // (1-z)h + z*h~
                }
            }
        }
        grid_barrier(bar, (++epoch) * NWG);
    }

    // ---------------- Final y_{T-1} from h after step T-1 (slot 0, T even) ----------------
    const float* hFin = ws;   // (T_ & 1) == 0
    for (int task = wave; task < 32; task += NWAVES) {
        const int n0 = task * 16;
        const int n  = n0 + l15;
        v8f c[4] = {};
        const float* bp = Wy + (size_t)n * H_ + koff;
        const float* a0 = hFin + (size_t)( 0 + l15) * H_ + koff;
        const float* a1 = hFin + (size_t)(16 + l15) * H_ + koff;
        const float* a2 = hFin + (size_t)(32 + l15) * H_ + koff;
        const float* a3 = hFin + (size_t)(48 + l15) * H_ + koff;
        mma_strip(c, a0, a1, a2, a3, bp, H_ / 4);
        const float bn = by[n];
#pragma unroll
        for (int j = 0; j < 4; ++j) {
#pragma unroll
            for (int i = 0; i < 8; ++i) {
                const int m = 16 * j + i + 8 * hi;
                out[(size_t)m * (T_ * E_) + (size_t)(T_ - 1) * E_ + n] = c[j][i] + bn;
            }
        }
    }
}

extern "C" void kernel_launch(void* const* d_in, const int* in_sizes, int n_in,
                              void* d_out, int out_size, void* d_ws, size_t ws_size,
                              hipStream_t stream)
{
    const float* word = (const float*)d_in[0];
    const float* Wz   = (const float*)d_in[1];
    const float* bz   = (const float*)d_in[2];
    const float* Wr   = (const float*)d_in[3];
    const float* br   = (const float*)d_in[4];
    const float* Wh   = (const float*)d_in[5];
    const float* bh   = (const float*)d_in[6];
    const float* Wy   = (const float*)d_in[7];
    const float* by   = (const float*)d_in[8];

    float*    ws  = (float*)d_ws;
    // layout: h[2][B*H] | z[B*H] | rh[B*H] | barrier counter
    unsigned* bar = (unsigned*)((char*)d_ws + (size_t)4 * B_ * H_ * sizeof(float));

    // Re-initialize h0 and barrier counter every call (deterministic, graph-safe).
    gru_init_ws<<<(B_ * H_ + 255) / 256, 256, 0, stream>>>(ws, bar);
    gru_persistent<<<NWG, BLK, 0, stream>>>(word, Wz, bz, Wr, br, Wh, bh, Wy, by,
                                            (float*)d_out, ws, bar);
}